// Grouper_35665408426483
// MI455X (gfx1250) — compile-verified
//
#include <hip/hip_runtime.h>
#include <stdint.h>

// Problem constants (from reference): xyz [8, 8192, 3] f32, S = N/4, K = 32
#define BATCH   8
#define NPTS    8192
#define NSAMP   2048
#define KNEIGH  32
#define CHUNK   128           // points per KNN chunk (8 waves x 16)
#define FPS_THREADS 1024

typedef __attribute__((ext_vector_type(2))) float v2f;
typedef __attribute__((ext_vector_type(8))) float v8f;

__device__ __forceinline__ unsigned long long umax64(unsigned long long a, unsigned long long b) { return a > b ? a : b; }
__device__ __forceinline__ unsigned long long umin64(unsigned long long a, unsigned long long b) { return a < b ? a : b; }

// Full bitonic sort (ascending) of 32 keys striped across the wave32 lanes.
__device__ __forceinline__ unsigned long long bitonic_sort32(unsigned long long x, int lane) {
#pragma unroll
  for (int k = 2; k <= 32; k <<= 1) {
#pragma unroll
    for (int j = k >> 1; j >= 1; j >>= 1) {
      unsigned long long v = __shfl_xor(x, j, 32);
      bool up    = ((lane & k) == 0);
      bool lower = ((lane & j) == 0);
      x = (lower == up) ? umin64(x, v) : umax64(x, v);
    }
  }
  return x;
}

// Re-sort a bitonic sequence ascending (5 stages).
__device__ __forceinline__ unsigned long long bitonic_merge32(unsigned long long x, int lane) {
#pragma unroll
  for (int j = 16; j >= 1; j >>= 1) {
    unsigned long long v = __shfl_xor(x, j, 32);
    x = ((lane & j) == 0) ? umin64(x, v) : umax64(x, v);
  }
  return x;
}

// run: ascending sorted top-32-so-far. candSorted: ascending sorted 32 new keys.
// Returns ascending sorted smallest-32 of the union.
__device__ __forceinline__ unsigned long long topk_merge(unsigned long long run,
                                                         unsigned long long candSorted,
                                                         int lane) {
  unsigned long long rev = __shfl_xor(candSorted, 31, 32);  // reverse across lanes
  return bitonic_merge32(umin64(run, rev), lane);           // min(asc, desc) is bitonic
}

// ---------------------------------------------------------------------------
// Kernel 1: farthest point sampling. One workgroup per batch; whole cloud in
// LDS (96KB xyz + 32KB min-dist fits CDNA5's 320KB/WGP LDS).
// Writes center coordinates straight into the centers region of d_out.
// ---------------------------------------------------------------------------
__global__ __launch_bounds__(FPS_THREADS)
void fps_kernel(const float* __restrict__ xyz, float* __restrict__ centers) {
  extern __shared__ unsigned char smem[];
  float* px   = (float*)smem;
  float* py   = px + NPTS;
  float* pz   = py + NPTS;
  float* mind = pz + NPTS;
  unsigned long long* red = (unsigned long long*)(mind + NPTS);  // 32 entries
  int* curSh = (int*)(red + 32);

  const int b    = blockIdx.x;
  const int tid  = threadIdx.x;
  const int lane = tid & 31;
  const int wave = tid >> 5;

  const float* base = xyz + (size_t)b * NPTS * 3;
  for (int i = tid; i < NPTS; i += FPS_THREADS) {
    px[i] = base[i * 3 + 0];
    py[i] = base[i * 3 + 1];
    pz[i] = base[i * 3 + 2];
    mind[i] = 3.402823466e38f;
  }
  __syncthreads();

  float* cout = centers + (size_t)b * NSAMP * 3;
  if (tid == 0) { cout[0] = px[0]; cout[1] = py[0]; cout[2] = pz[0]; }

  int cur = 0;
  for (int s = 1; s < NSAMP; ++s) {
    float cx = px[cur], cy = py[cur], cz = pz[cur];
    unsigned long long best = 0ull;
#pragma unroll
    for (int r = 0; r < NPTS / FPS_THREADS; ++r) {
      int i = tid + r * FPS_THREADS;
      float dx = px[i] - cx, dy = py[i] - cy, dz = pz[i] - cz;
      float d = dx * dx + dy * dy + dz * dz;
      float m = fminf(mind[i], d);
      mind[i] = m;
      // max key -> max dist; ties -> larger (N-1-i) -> smaller i (argmax semantics)
      unsigned long long key =
          ((unsigned long long)__float_as_uint(m) << 32) | (unsigned)(NPTS - 1 - i);
      best = umax64(best, key);
    }
#pragma unroll
    for (int off = 16; off >= 1; off >>= 1)
      best = umax64(best, __shfl_xor(best, off, 32));
    if (lane == 0) red[wave] = best;
    __syncthreads();
    if (wave == 0) {
      unsigned long long v = red[lane];
#pragma unroll
      for (int off = 16; off >= 1; off >>= 1)
        v = umax64(v, __shfl_xor(v, off, 32));
      if (lane == 0) {
        int idx = NPTS - 1 - (int)(v & 0xFFFFFFFFull);
        *curSh = idx;
        cout[s * 3 + 0] = px[idx];
        cout[s * 3 + 1] = py[idx];
        cout[s * 3 + 2] = pz[idx];
      }
    }
    __syncthreads();
    cur = *curSh;
  }
}

// ---------------------------------------------------------------------------
// Kernel 2: exact KNN (K=32) + gather + re-center.
// One 256-thread workgroup per (batch, 16-center tile). Per 128-point chunk,
// each of the 8 waves computes one 16x16 distance tile with
// V_WMMA_F32_16X16X4_F32:  A row m = (-2cx,-2cy,-2cz, 1), B col n =
// (px,py,pz,|p|^2)  =>  D[m][n] = |p|^2 - 2 c.p ; add |c|^2 per row.
// Selection: wave-level exact bitonic top-32 (2 centers per wave), with a
// wave-uniform threshold skip: if no candidate beats the current 32nd-best
// key (lane 31 of the sorted run), the whole sort+merge is an identity and
// is skipped.
// ---------------------------------------------------------------------------
__global__ __launch_bounds__(256)
void knn_kernel(const float* __restrict__ xyz, const float* __restrict__ centers,
                float* __restrict__ nbrOut) {
  __shared__ float d2buf[16 * CHUNK];  // 8 KB distance tile
  __shared__ float cc[16 * 3];         // 16 center coords

  const int blk  = blockIdx.x;
  const int b    = blk >> 7;    // / (NSAMP/16)
  const int tile = blk & 127;
  const int tid  = threadIdx.x;
  const int lane = tid & 31;
  const int wave = tid >> 5;    // 0..7

  const float* xb = xyz + (size_t)b * NPTS * 3;
  const float* cb = centers + ((size_t)b * NSAMP + (size_t)tile * 16) * 3;

  if (tid < 48) cc[tid] = cb[tid];
  __syncthreads();

  // A-matrix fragment (identical in every wave). 16x4 f32 layout:
  // lanes 0-15: (K=0,K=1) of row M=lane ; lanes 16-31: (K=2,K=3) of row M=lane-16.
  const int mrow = lane & 15;
  v2f amat;
  {
    float cx = cc[mrow * 3 + 0], cy = cc[mrow * 3 + 1], cz = cc[mrow * 3 + 2];
    if (lane < 16) { amat.x = -2.0f * cx; amat.y = -2.0f * cy; }
    else           { amat.x = -2.0f * cz; amat.y = 1.0f; }
  }
  // Per-accumulator-register center-norm addend (C/D row = r or r+8 by lane half).
  float cnr[8];
#pragma unroll
  for (int r = 0; r < 8; ++r) {
    int row = (lane < 16) ? r : (r + 8);
    float x = cc[row * 3 + 0], y = cc[row * 3 + 1], z = cc[row * 3 + 2];
    cnr[r] = x * x + y * y + z * z;
  }

  unsigned long long run0 = ~0ull, run1 = ~0ull;        // sorted top-32 lists
  unsigned long long runMax0 = ~0ull, runMax1 = ~0ull;  // lane-31 (worst kept) key
  const int c0 = 2 * wave, c1 = 2 * wave + 1;

  for (int chunk = 0; chunk < NPTS; chunk += CHUNK) {
    // ---- distance tile (wave w covers points [chunk + 16w, +16)) ----
    const int p = chunk + wave * 16 + (lane & 15);
    float pxv = xb[p * 3 + 0], pyv = xb[p * 3 + 1], pzv = xb[p * 3 + 2];
    if (chunk + CHUNK < NPTS)
      __builtin_prefetch(xb + (size_t)(p + CHUNK) * 3, 0, 1);

    const int col = wave * 16 + (lane & 15);
#if __has_builtin(__builtin_amdgcn_wmma_f32_16x16x4_f32)
    v2f bmat;
    if (lane < 16) { bmat.x = pxv; bmat.y = pyv; }
    else           { bmat.x = pzv; bmat.y = pxv * pxv + pyv * pyv + pzv * pzv; }
    v8f acc = {};
    acc = __builtin_amdgcn_wmma_f32_16x16x4_f32(
        /*neg_a=*/false, amat, /*neg_b=*/false, bmat,
        /*c_mod=*/(short)0, acc, /*reuse_a=*/false, /*reuse_b=*/false);
#pragma unroll
    for (int r = 0; r < 8; ++r) {
      int row = (lane < 16) ? r : (r + 8);
      d2buf[row * CHUNK + col] = fmaxf(acc[r] + cnr[r], 0.0f);
    }
#else
    // Scalar fallback (same LDS layout): lane pair (l, l+16) share a point.
    float pn = pxv * pxv + pyv * pyv + pzv * pzv;
#pragma unroll
    for (int r = 0; r < 8; ++r) {
      int row = (lane < 16) ? r : (r + 8);
      float dot = cc[row * 3 + 0] * pxv + cc[row * 3 + 1] * pyv + cc[row * 3 + 2] * pzv;
      d2buf[row * CHUNK + col] = fmaxf(pn - 2.0f * dot + cnr[r], 0.0f);
    }
#endif
    __syncthreads();

    // ---- exact top-32 merge: each wave owns centers c0, c1 ----
#pragma unroll
    for (int t = 0; t < 4; ++t) {
      unsigned pidx = (unsigned)(chunk + t * 32 + lane);
      {
        unsigned d = __float_as_uint(d2buf[c0 * CHUNK + t * 32 + lane]);
        unsigned long long key = ((unsigned long long)d << 32) | pidx;
        if (__any(key < runMax0)) {            // wave-uniform: merge not identity
          key  = bitonic_sort32(key, lane);
          run0 = topk_merge(run0, key, lane);
          runMax0 = __shfl(run0, 31, 32);
        }
      }
      {
        unsigned d = __float_as_uint(d2buf[c1 * CHUNK + t * 32 + lane]);
        unsigned long long key = ((unsigned long long)d << 32) | pidx;
        if (__any(key < runMax1)) {
          key  = bitonic_sort32(key, lane);
          run1 = topk_merge(run1, key, lane);
          runMax1 = __shfl(run1, 31, 32);
        }
      }
    }
    __syncthreads();
  }

  // ---- gather + re-center: lane i writes neighbor i of each owned center ----
#pragma unroll
  for (int h = 0; h < 2; ++h) {
    int c = (h == 0) ? c0 : c1;
    unsigned long long run = (h == 0) ? run0 : run1;
    float ccx = cc[c * 3 + 0], ccy = cc[c * 3 + 1], ccz = cc[c * 3 + 2];
    int idx = (int)(run & 0xFFFFFFFFull);
    const float* pp = xb + (size_t)idx * 3;
    size_t o = ((((size_t)b * NSAMP + (size_t)tile * 16 + c) * KNEIGH) + lane) * 3;
    nbrOut[o + 0] = pp[0] - ccx;
    nbrOut[o + 1] = pp[1] - ccy;
    nbrOut[o + 2] = pp[2] - ccz;
  }
}

extern "C" void kernel_launch(void* const* d_in, const int* in_sizes, int n_in,
                              void* d_out, int out_size, void* d_ws, size_t ws_size,
                              hipStream_t stream) {
  (void)in_sizes; (void)n_in; (void)out_size; (void)d_ws; (void)ws_size;
  const float* xyz = (const float*)d_in[0];
  float* out = (float*)d_out;
  // Output layout: neighborhood [B,S,K,3] then centers [B,S,3]
  float* centers = out + (size_t)BATCH * NSAMP * KNEIGH * 3;

  size_t fps_smem = (size_t)NPTS * 4 * sizeof(float)         // px,py,pz,mind
                  + 32 * sizeof(unsigned long long)          // wave reduction
                  + 16;                                      // cur broadcast
  fps_kernel<<<BATCH, FPS_THREADS, fps_smem, stream>>>(xyz, centers);

  knn_kernel<<<BATCH * (NSAMP / 16), 256, 0, stream>>>(xyz, centers, out);
}